// GATClassifier_64372969832849
// MI455X (gfx1250) — compile-verified
//
#include <hip/hip_runtime.h>
#include <math.h>

#define NN    50000
#define EE    400000
#define ETOT  (EE + NN)
#define GG    64
#define FIN   16
#define HID   128
#define HEADS 4
#define C1    (HEADS * HID)   // 512
#define GFD   10
#define NCLS  6
#define APAD  516             // LDS row stride (floats) for GEMM2 A tile

typedef float v2f __attribute__((ext_vector_type(2)));
typedef float v8f __attribute__((ext_vector_type(8)));

__device__ __forceinline__ v8f wmma_k4(v2f a, v2f b, v8f c) {
  // (neg_a, A, neg_b, B, c_mod, C, reuse_a, reuse_b)
  return __builtin_amdgcn_wmma_f32_16x16x4_f32(false, a, false, b, (short)0, c,
                                               false, false);
}

// ---------------- GEMM1: h1[N,512] = x[N,16] @ W1[16,512] ----------------
__global__ void k_gemm1(const float* __restrict__ x, const float* __restrict__ W1,
                        float* __restrict__ h1) {
  int lane = threadIdx.x & 31;
  int wave = threadIdx.x >> 5;  // 0..7
  int lid  = lane & 15;
  int half = lane >> 4;
  int rowbase = blockIdx.x * 16;
  const float* xrow = x + (size_t)(rowbase + lid) * FIN;
#pragma unroll
  for (int ct = 0; ct < 4; ++ct) {
    int colbase = (wave * 4 + ct) * 16;
    v8f c = {};
#pragma unroll
    for (int k0 = 0; k0 < FIN; k0 += 4) {
      int k = k0 + 2 * half;
      v2f a, b;
      a[0] = xrow[k];
      a[1] = xrow[k + 1];
      b[0] = W1[(size_t)k * C1 + colbase + lid];
      b[1] = W1[(size_t)(k + 1) * C1 + colbase + lid];
      c = wmma_k4(a, b, c);
    }
    float* op = h1 + (size_t)rowbase * C1 + colbase + lid;
#pragma unroll
    for (int r = 0; r < 8; ++r) op[(size_t)(r + 8 * half) * C1] = c[r];
  }
}

// ---------------- attention logits, layer 1: a[n,h] = dot(h1[n,h,:], att[h,:]) ----
__global__ void k_att1(const float* __restrict__ h1, const float* __restrict__ attS,
                       const float* __restrict__ attD, float* __restrict__ aS,
                       float* __restrict__ aD) {
  int t = blockIdx.x * blockDim.x + threadIdx.x;
  if (t >= NN * HEADS) return;
  int n = t >> 2, h = t & 3;
  const float4* row = (const float4*)(h1 + (size_t)n * C1 + h * HID);
  const float4* pa  = (const float4*)(attS + h * HID);
  const float4* pb  = (const float4*)(attD + h * HID);
  float s = 0.f, d = 0.f;
#pragma unroll 4
  for (int i = 0; i < HID / 4; ++i) {
    float4 v = row[i], a = pa[i], b = pb[i];
    s += v.x * a.x + v.y * a.y + v.z * a.z + v.w * a.w;
    d += v.x * b.x + v.y * b.y + v.z * b.z + v.w * b.w;
  }
  aS[t] = s; aD[t] = d;
}

// ---------------- attention logits, layer 2 (1 head) ----------------
__global__ void k_att2(const float* __restrict__ hp, const float* __restrict__ attS,
                       const float* __restrict__ attD, float* __restrict__ aS,
                       float* __restrict__ aD) {
  int n = blockIdx.x * blockDim.x + threadIdx.x;
  if (n >= NN) return;
  const float4* row = (const float4*)(hp + (size_t)n * HID);
  const float4* pa  = (const float4*)attS;
  const float4* pb  = (const float4*)attD;
  float s = 0.f, d = 0.f;
#pragma unroll 4
  for (int i = 0; i < HID / 4; ++i) {
    float4 v = row[i], a = pa[i], b = pb[i];
    s += v.x * a.x + v.y * a.y + v.z * a.z + v.w * a.w;
    d += v.x * b.x + v.y * b.y + v.z * b.z + v.w * b.w;
  }
  aS[n] = s; aD[n] = d;
}

// ---------------- CSR build (dst-major), self-loop included via deg init ----
__global__ void k_deg_init(int* __restrict__ deg) {
  int n = blockIdx.x * blockDim.x + threadIdx.x;
  if (n < NN) deg[n] = 1;  // self loop
}
__global__ void k_deg_count(const int* __restrict__ ei, int* __restrict__ deg) {
  int e = blockIdx.x * blockDim.x + threadIdx.x;
  if (e < EE) atomicAdd(&deg[ei[EE + e]], 1);
}
__global__ void k_scan(const int* __restrict__ deg, int* __restrict__ rowptr,
                       int* __restrict__ cursor) {
  __shared__ int sm[1024];
  __shared__ int carry;
  if (threadIdx.x == 0) carry = 0;
  __syncthreads();
  for (int base = 0; base < NN; base += 1024) {
    int i = base + threadIdx.x;
    int v = (i < NN) ? deg[i] : 0;
    sm[threadIdx.x] = v;
    __syncthreads();
    for (int off = 1; off < 1024; off <<= 1) {
      int t = (threadIdx.x >= off) ? sm[threadIdx.x - off] : 0;
      __syncthreads();
      sm[threadIdx.x] += t;
      __syncthreads();
    }
    int excl = sm[threadIdx.x] - v + carry;
    if (i < NN) { rowptr[i] = excl; cursor[i] = excl; }
    __syncthreads();
    if (threadIdx.x == 1023) carry += sm[1023];
    __syncthreads();
  }
  if (threadIdx.x == 0) rowptr[NN] = carry;  // == ETOT
}
__global__ void k_fill(const int* __restrict__ ei, int* __restrict__ cursor,
                       int* __restrict__ colsrc) {
  int t = blockIdx.x * blockDim.x + threadIdx.x;
  if (t < EE) {
    int s = ei[t], d = ei[EE + t];
    int p = atomicAdd(&cursor[d], 1);
    colsrc[p] = s;
  } else if (t < ETOT) {
    int n = t - EE;
    int p = atomicAdd(&cursor[n], 1);
    colsrc[p] = n;  // self loop
  }
}

// ---------------- GAT aggregation layer 1: wave per node, 4 heads x 128 ----
__global__ void k_agg1(const float* __restrict__ h1, const float* __restrict__ aS,
                       const float* __restrict__ aD, const int* __restrict__ rowptr,
                       const int* __restrict__ colsrc, const float* __restrict__ bias,
                       float* __restrict__ out) {
  int lane = threadIdx.x & 31;
  int node = blockIdx.x * 8 + (threadIdx.x >> 5);
  if (node >= NN) return;
  int rs = rowptr[node], re = rowptr[node + 1];
  float ad[4];
#pragma unroll
  for (int h = 0; h < 4; ++h) ad[h] = aD[node * 4 + h];
  float m[4] = {-1e30f, -1e30f, -1e30f, -1e30f};
  for (int j = rs + lane; j < re; j += 32) {
    int s = colsrc[j];
#pragma unroll
    for (int h = 0; h < 4; ++h) {
      float e = aS[s * 4 + h] + ad[h];
      e = e > 0.f ? e : 0.2f * e;
      m[h] = fmaxf(m[h], e);
    }
  }
#pragma unroll
  for (int off = 16; off; off >>= 1)
#pragma unroll
    for (int h = 0; h < 4; ++h) m[h] = fmaxf(m[h], __shfl_xor(m[h], off, 32));
  float den[4] = {0.f, 0.f, 0.f, 0.f};
  for (int j = rs + lane; j < re; j += 32) {
    int s = colsrc[j];
#pragma unroll
    for (int h = 0; h < 4; ++h) {
      float e = aS[s * 4 + h] + ad[h];
      e = e > 0.f ? e : 0.2f * e;
      den[h] += expf(e - m[h]);
    }
  }
#pragma unroll
  for (int off = 16; off; off >>= 1)
#pragma unroll
    for (int h = 0; h < 4; ++h) den[h] += __shfl_xor(den[h], off, 32);
  // this lane owns channels [lane*16, lane*16+16) -> head = lane/8
  int hs = lane >> 3;
  float adL = hs == 0 ? ad[0] : hs == 1 ? ad[1] : hs == 2 ? ad[2] : ad[3];
  float mL  = hs == 0 ? m[0]  : hs == 1 ? m[1]  : hs == 2 ? m[2]  : m[3];
  float dL  = hs == 0 ? den[0]: hs == 1 ? den[1]: hs == 2 ? den[2]: den[3];
  float invd = 1.f / dL;
  float acc[16];
#pragma unroll
  for (int i = 0; i < 16; ++i) acc[i] = 0.f;
  for (int j = rs; j < re; ++j) {
    int s = colsrc[j];
    float e = aS[s * 4 + hs] + adL;
    e = e > 0.f ? e : 0.2f * e;
    float w = expf(e - mL) * invd;
    const float4* hv = (const float4*)(h1 + (size_t)s * C1 + lane * 16);
#pragma unroll
    for (int q = 0; q < 4; ++q) {
      float4 v = hv[q];
      acc[q * 4 + 0] += w * v.x; acc[q * 4 + 1] += w * v.y;
      acc[q * 4 + 2] += w * v.z; acc[q * 4 + 3] += w * v.w;
    }
  }
  const float* bb = bias + lane * 16;
  float* op = out + (size_t)node * C1 + lane * 16;
#pragma unroll
  for (int q = 0; q < 4; ++q) {
    float4 r;
    float v0 = acc[q * 4 + 0] + bb[q * 4 + 0]; r.x = v0 > 0.f ? v0 : expf(v0) - 1.f;
    float v1 = acc[q * 4 + 1] + bb[q * 4 + 1]; r.y = v1 > 0.f ? v1 : expf(v1) - 1.f;
    float v2 = acc[q * 4 + 2] + bb[q * 4 + 2]; r.z = v2 > 0.f ? v2 : expf(v2) - 1.f;
    float v3 = acc[q * 4 + 3] + bb[q * 4 + 3]; r.w = v3 > 0.f ? v3 : expf(v3) - 1.f;
    *(float4*)(op + q * 4) = r;
  }
}

// ---------------- GEMM2: hpre2[N,128] = h1a[N,512] @ W2[512,128] ----------
__global__ void k_gemm2(const float* __restrict__ A, const float* __restrict__ W2,
                        float* __restrict__ out) {
  __shared__ float As[16 * APAD];
  int lane = threadIdx.x & 31;
  int wave = threadIdx.x >> 5;  // 0..7 -> column tile
  int lid  = lane & 15;
  int half = lane >> 4;
  int rowbase = blockIdx.x * 16;
  // cooperative load of contiguous 16x512 A tile into padded LDS
  const float4* src = (const float4*)(A + (size_t)rowbase * C1);
  for (int f = threadIdx.x; f < 16 * 128; f += 256) {
    int r = f >> 7, c4 = f & 127;
    *(float4*)(&As[r * APAD + c4 * 4]) = src[f];
  }
  __syncthreads();
  int colbase = wave * 16;
  const float* Bp = W2 + colbase + lid;
  const float* Ap = As + lid * APAD;
  v8f c = {};
#pragma unroll 4
  for (int k0 = 0; k0 < C1; k0 += 4) {
    int k = k0 + 2 * half;
    v2f a, b;
    a[0] = Ap[k];
    a[1] = Ap[k + 1];
    b[0] = Bp[(size_t)k * HID];
    b[1] = Bp[(size_t)(k + 1) * HID];
    c = wmma_k4(a, b, c);
  }
  float* op = out + (size_t)rowbase * HID + colbase + lid;
#pragma unroll
  for (int r = 0; r < 8; ++r) op[(size_t)(r + 8 * half) * HID] = c[r];
}

// ---------------- GAT aggregation layer 2: wave per node, 1 head x 128 ----
__global__ void k_agg2(const float* __restrict__ hp, const float* __restrict__ aS,
                       const float* __restrict__ aD, const int* __restrict__ rowptr,
                       const int* __restrict__ colsrc, const float* __restrict__ bias,
                       float* __restrict__ out) {
  int lane = threadIdx.x & 31;
  int node = blockIdx.x * 8 + (threadIdx.x >> 5);
  if (node >= NN) return;
  int rs = rowptr[node], re = rowptr[node + 1];
  float ad = aD[node];
  float m = -1e30f;
  for (int j = rs + lane; j < re; j += 32) {
    float e = aS[colsrc[j]] + ad;
    e = e > 0.f ? e : 0.2f * e;
    m = fmaxf(m, e);
  }
#pragma unroll
  for (int off = 16; off; off >>= 1) m = fmaxf(m, __shfl_xor(m, off, 32));
  float den = 0.f;
  for (int j = rs + lane; j < re; j += 32) {
    float e = aS[colsrc[j]] + ad;
    e = e > 0.f ? e : 0.2f * e;
    den += expf(e - m);
  }
#pragma unroll
  for (int off = 16; off; off >>= 1) den += __shfl_xor(den, off, 32);
  float invd = 1.f / den;
  float a0 = 0.f, a1 = 0.f, a2 = 0.f, a3 = 0.f;
  for (int j = rs; j < re; ++j) {
    int s = colsrc[j];
    float e = aS[s] + ad;
    e = e > 0.f ? e : 0.2f * e;
    float w = expf(e - m) * invd;
    float4 v = *(const float4*)(hp + (size_t)s * HID + lane * 4);
    a0 += w * v.x; a1 += w * v.y; a2 += w * v.z; a3 += w * v.w;
  }
  const float* bb = bias + lane * 4;
  float4 r;
  float v0 = a0 + bb[0]; r.x = v0 > 0.f ? v0 : expf(v0) - 1.f;
  float v1 = a1 + bb[1]; r.y = v1 > 0.f ? v1 : expf(v1) - 1.f;
  float v2 = a2 + bb[2]; r.z = v2 > 0.f ? v2 : expf(v2) - 1.f;
  float v3 = a3 + bb[3]; r.w = v3 > 0.f ? v3 : expf(v3) - 1.f;
  *(float4*)(out + (size_t)node * HID + lane * 4) = r;
}

// ---------------- global mean pool ----------------
__global__ void k_pool_init(float* __restrict__ sums, float* __restrict__ counts) {
  int t = blockIdx.x * blockDim.x + threadIdx.x;
  if (t < GG * HID) sums[t] = 0.f;
  if (t < GG) counts[t] = 0.f;
}
__global__ void k_pool(const float* __restrict__ h2, const int* __restrict__ batch,
                       float* __restrict__ sums, float* __restrict__ counts) {
  int t = blockIdx.x * blockDim.x + threadIdx.x;
  if (t >= NN * 32) return;
  int n = t >> 5, q = t & 31;
  int b = batch[n];
  float4 v = *(const float4*)(h2 + (size_t)n * HID + q * 4);
  float* s = sums + (size_t)b * HID + q * 4;
  atomicAdd(s + 0, v.x); atomicAdd(s + 1, v.y);
  atomicAdd(s + 2, v.z); atomicAdd(s + 3, v.w);
  if (q == 0) atomicAdd(&counts[b], 1.0f);
}

// ---------------- head: graph MLP + classifier, one block per graph --------
__global__ void k_head(const float* __restrict__ sums, const float* __restrict__ counts,
                       const float* __restrict__ gfeat,
                       const float* __restrict__ Wg1, const float* __restrict__ bg1,
                       const float* __restrict__ Wg2, const float* __restrict__ bg2,
                       const float* __restrict__ Wc1, const float* __restrict__ bc1,
                       const float* __restrict__ Wc2, const float* __restrict__ bc2,
                       float* __restrict__ out) {
  __shared__ float z[HID + 32];   // [pooled(128) | gf(32)]
  __shared__ float t32[32];
  __shared__ float z1[HID];
  int g = blockIdx.x, t = threadIdx.x;
  if (t < HID) {
    float cnt = counts[g];
    z[t] = sums[(size_t)g * HID + t] / fmaxf(cnt, 1.0f);
  }
  if (t < 32) {
    float s = bg1[t];
    for (int k = 0; k < GFD; ++k) s += gfeat[g * GFD + k] * Wg1[k * 32 + t];
    t32[t] = fmaxf(s, 0.f);
  }
  __syncthreads();
  if (t < 32) {
    float s = bg2[t];
    for (int k = 0; k < 32; ++k) s += t32[k] * Wg2[k * 32 + t];
    z[HID + t] = s;
  }
  __syncthreads();
  if (t < HID) {
    float s = bc1[t];
    for (int k = 0; k < HID + 32; ++k) s += z[k] * Wc1[k * HID + t];
    z1[t] = fmaxf(s, 0.f);
  }
  __syncthreads();
  if (t < NCLS) {
    float s = bc2[t];
    for (int k = 0; k < HID; ++k) s += z1[k] * Wc2[k * NCLS + t];
    out[g * NCLS + t] = s;
  }
}

extern "C" void kernel_launch(void* const* d_in, const int* in_sizes, int n_in,
                              void* d_out, int out_size, void* d_ws, size_t ws_size,
                              hipStream_t stream) {
  const float* x     = (const float*)d_in[0];
  const int*   ei    = (const int*)d_in[1];
  const int*   batch = (const int*)d_in[2];
  const float* gfeat = (const float*)d_in[3];
  const float* W1    = (const float*)d_in[4];
  const float* as1   = (const float*)d_in[5];
  const float* ad1   = (const float*)d_in[6];
  const float* b1    = (const float*)d_in[7];
  const float* W2    = (const float*)d_in[8];
  const float* as2   = (const float*)d_in[9];
  const float* ad2   = (const float*)d_in[10];
  const float* b2    = (const float*)d_in[11];
  const float* Wg1   = (const float*)d_in[12];
  const float* bg1   = (const float*)d_in[13];
  const float* Wg2   = (const float*)d_in[14];
  const float* bg2   = (const float*)d_in[15];
  const float* Wc1   = (const float*)d_in[16];
  const float* bc1   = (const float*)d_in[17];
  const float* Wc2   = (const float*)d_in[18];
  const float* bc2   = (const float*)d_in[19];

  char* ws = (char*)d_ws;
  size_t off = 0;
  auto take = [&](size_t bytes) -> char* {
    char* p = ws + off;
    off = (off + bytes + 255) & ~(size_t)255;
    return p;
  };
  float* h1     = (float*)take((size_t)NN * C1 * 4);
  float* h1a    = (float*)take((size_t)NN * C1 * 4);
  float* aS1    = (float*)take((size_t)NN * HEADS * 4);
  float* aD1    = (float*)take((size_t)NN * HEADS * 4);
  float* aS2    = (float*)take((size_t)NN * 4);
  float* aD2    = (float*)take((size_t)NN * 4);
  int*   deg    = (int*)take((size_t)NN * 4);
  int*   rowptr = (int*)take((size_t)(NN + 1) * 4);
  int*   cursor = (int*)take((size_t)NN * 4);
  int*   colsrc = (int*)take((size_t)ETOT * 4);
  float* sums   = (float*)take((size_t)GG * HID * 4);
  float* counts = (float*)take((size_t)GG * 4);
  float* hpre2  = h1;    // reuse: h1 dead after k_agg1
  float* h2     = h1a;   // reuse: h1a dead after k_gemm2

  // ---- layer 1 ----
  k_gemm1<<<NN / 16, 256, 0, stream>>>(x, W1, h1);
  k_att1<<<(NN * HEADS + 255) / 256, 256, 0, stream>>>(h1, as1, ad1, aS1, aD1);
  // ---- CSR (shared by both layers) ----
  k_deg_init<<<(NN + 255) / 256, 256, 0, stream>>>(deg);
  k_deg_count<<<(EE + 255) / 256, 256, 0, stream>>>(ei, deg);
  k_scan<<<1, 1024, 0, stream>>>(deg, rowptr, cursor);
  k_fill<<<(ETOT + 255) / 256, 256, 0, stream>>>(ei, cursor, colsrc);
  k_agg1<<<(NN + 7) / 8, 256, 0, stream>>>(h1, aS1, aD1, rowptr, colsrc, b1, h1a);
  // ---- layer 2 ----
  k_gemm2<<<NN / 16, 256, 0, stream>>>(h1a, W2, hpre2);
  k_att2<<<(NN + 255) / 256, 256, 0, stream>>>(hpre2, as2, ad2, aS2, aD2);
  k_agg2<<<(NN + 7) / 8, 256, 0, stream>>>(hpre2, aS2, aD2, rowptr, colsrc, b2, h2);
  // ---- pool + head ----
  k_pool_init<<<(GG * HID + 255) / 256, 256, 0, stream>>>(sums, counts);
  k_pool<<<(NN * 32 + 255) / 256, 256, 0, stream>>>(h2, batch, sums, counts);
  k_head<<<GG, 160, 0, stream>>>(sums, counts, gfeat, Wg1, bg1, Wg2, bg2,
                                 Wc1, bc1, Wc2, bc2, (float*)d_out);
}